// DEEP_GATGNN_18726057411353
// MI455X (gfx1250) — compile-verified
//
#include <hip/hip_runtime.h>
#include <math.h>

// DeeperGATGNN forward for MI455X (gfx1250, wave32).
// Heavy GEMMs use V_WMMA_F32_16X16X4_F32 (exact f32 matrix path).
// Workspace requirement: ~234 MB (58.5M floats).

#define NNODES 10000
#define NEDGES 160000
#define NGRAPH 64
#define DIM    64
#define HEADS  4
#define HD     256     // HEADS*DIM
#define GROUPS 10
#define NLAYER 5

typedef float v2f __attribute__((ext_vector_type(2)));
typedef float v8f __attribute__((ext_vector_type(8)));

__device__ __forceinline__ float sp_f(float x) {           // stable softplus = logaddexp(x,0)
  return fmaxf(x, 0.f) + log1pf(expf(-fabsf(x)));
}
__device__ __forceinline__ float lrelu(float x) { return x > 0.f ? x : 0.2f * x; }

// ---------------------------------------------------------------------------
// WMMA f32 GEMM: C[M,Nn] = act(A[M,K] @ B[K,Nn] + bias)
// One wave32 owns a 16x64 output strip: 4 accumulators, A fragment reused
// across the 4 column tiles. Requires M%16==0 and Nn%64==0 (true for all call
// sites); K arbitrary.
// Main K loop is guard-free (no divergent exec-masked loads); the K%4 tail is
// handled once with clamped addresses + selects (v_cndmask, not branches).
// Fragment layout per CDNA5 ISA 7.12.2:
//   A 16x4 f32: VGPR0 = K=0 (lanes0-15) / K=2 (lanes16-31); VGPR1 = K=1 / K=3
//   B 4x16 f32: mirrored (N striped across lanes within a VGPR)
//   C/D 16x16:  VGPR v -> M=v (lanes0-15) / M=v+8 (lanes16-31), N=lane&15
// ---------------------------------------------------------------------------
__device__ __forceinline__ void store_tile(float* __restrict__ C, int ldc,
                                           int mrow, int c, v8f acc,
                                           const float* __restrict__ bias, int act)
{
  const float bv = bias ? bias[c] : 0.f;
  #pragma unroll
  for (int v = 0; v < 8; ++v) {
    float val = acc[v] + bv;
    if      (act == 1) val = lrelu(val);
    else if (act == 2) val = sp_f(val);
    C[(size_t)(mrow + v) * ldc + c] = val;
  }
}

__global__ __launch_bounds__(256)
void gemm_wmma_f32(const float* __restrict__ A, int lda,
                   const float* __restrict__ B, int ldb,
                   float* __restrict__ C, int ldc,
                   int M, int K, int Nn,
                   const float* __restrict__ bias, int act)
{
  const int lane  = threadIdx.x & 31;
  const int gwave = (int)((blockIdx.x * blockDim.x + threadIdx.x) >> 5);
  const int ngrp  = Nn >> 6;
  const int tm    = gwave / ngrp;
  const int tg    = gwave - tm * ngrp;
  if (tm * 16 >= M) return;
  const int r     = tm * 16 + (lane & 15);
  const int cbase = tg * 64 + (lane & 15);
  const int khi   = (lane >> 4) << 1;          // 0 (lanes 0-15) or 2 (lanes 16-31)
  v8f acc0 = {}; v8f acc1 = {}; v8f acc2 = {}; v8f acc3 = {};

  const float* __restrict__ Aptr = A + (size_t)r * lda + khi;       // -> {ka, ka+1}
  const float* __restrict__ Bptr = B + (size_t)khi * ldb + cbase;   // row ka, col base
  const size_t bstep = 4 * (size_t)ldb;
  const int Kmain = K & ~3;

  for (int k0 = 0; k0 < Kmain; k0 += 4) {
    v2f a;
    a.x = Aptr[0];
    a.y = Aptr[1];
    const float* __restrict__ B0 = Bptr;
    const float* __restrict__ B1 = Bptr + ldb;
    v2f b0, b1, b2, b3;
    b0.x = B0[0];  b0.y = B1[0];
    b1.x = B0[16]; b1.y = B1[16];
    b2.x = B0[32]; b2.y = B1[32];
    b3.x = B0[48]; b3.y = B1[48];
    acc0 = __builtin_amdgcn_wmma_f32_16x16x4_f32(false, a, false, b0, (short)0, acc0, false, false);
    acc1 = __builtin_amdgcn_wmma_f32_16x16x4_f32(false, a, false, b1, (short)0, acc1, false, false);
    acc2 = __builtin_amdgcn_wmma_f32_16x16x4_f32(false, a, false, b2, (short)0, acc2, false, false);
    acc3 = __builtin_amdgcn_wmma_f32_16x16x4_f32(false, a, false, b3, (short)0, acc3, false, false);
    Aptr += 4;
    Bptr += bstep;
  }

  if (Kmain < K) {           // tail (K%4 in {1,2,3}): clamped loads + selects
    const int ka  = Kmain + khi;
    const bool ok0 = (ka < K), ok1 = (ka + 1 < K);
    const int o0 = (ok0 ? ka : (K - 1)) - Kmain - khi;   // offsets rel. to Aptr
    const int o1 = (ok1 ? ka + 1 : (K - 1)) - Kmain - khi;
    v2f a;
    const float a0 = Aptr[o0], a1 = Aptr[o1];
    a.x = ok0 ? a0 : 0.f;
    a.y = ok1 ? a1 : 0.f;
    const float* __restrict__ B0 = Bptr + (size_t)o0 * ldb;
    const float* __restrict__ B1 = Bptr + (size_t)o1 * ldb;
    v2f b0, b1, b2, b3;
    const float t00 = B0[0],  t10 = B1[0];
    const float t01 = B0[16], t11 = B1[16];
    const float t02 = B0[32], t12 = B1[32];
    const float t03 = B0[48], t13 = B1[48];
    b0.x = ok0 ? t00 : 0.f;  b0.y = ok1 ? t10 : 0.f;
    b1.x = ok0 ? t01 : 0.f;  b1.y = ok1 ? t11 : 0.f;
    b2.x = ok0 ? t02 : 0.f;  b2.y = ok1 ? t12 : 0.f;
    b3.x = ok0 ? t03 : 0.f;  b3.y = ok1 ? t13 : 0.f;
    acc0 = __builtin_amdgcn_wmma_f32_16x16x4_f32(false, a, false, b0, (short)0, acc0, false, false);
    acc1 = __builtin_amdgcn_wmma_f32_16x16x4_f32(false, a, false, b1, (short)0, acc1, false, false);
    acc2 = __builtin_amdgcn_wmma_f32_16x16x4_f32(false, a, false, b2, (short)0, acc2, false, false);
    acc3 = __builtin_amdgcn_wmma_f32_16x16x4_f32(false, a, false, b3, (short)0, acc3, false, false);
  }

  const int mrow = tm * 16 + ((lane >> 4) << 3);
  store_tile(C, ldc, mrow, cbase,      acc0, bias, act);
  store_tile(C, ldc, mrow, cbase + 16, acc1, bias, act);
  store_tile(C, ldc, mrow, cbase + 32, acc2, bias, act);
  store_tile(C, ldc, mrow, cbase + 48, acc3, bias, act);
}

// ---------------------------------------------------------------------------
__global__ void zero_f32(float* __restrict__ p, long n) {
  long i = (long)blockIdx.x * blockDim.x + threadIdx.x;
  const long stride = (long)gridDim.x * blockDim.x;
  for (; i < n; i += stride) p[i] = 0.f;
}

// Per-edge: h_i = sp(P[src]+Q), h_j = sp(P[dst]+Q) (h_j overwrites Q in place),
// alpha_raw[e,h] = sp( sum_d h_i*att[h,d] + h_j*att[h,64+d] ). One wave/edge.
__global__ void edge_msg(const float* __restrict__ P, float* __restrict__ QHj,
                         const int* __restrict__ src, const int* __restrict__ dst,
                         const float* __restrict__ att, float* __restrict__ alpha_raw)
{
  const int e    = (int)((blockIdx.x * blockDim.x + threadIdx.x) >> 5);
  const int lane = threadIdx.x & 31;
  if (e >= NEDGES) return;
  const float* Ps = P + (size_t)src[e] * HD;
  const float* Pd = P + (size_t)dst[e] * HD;
  float*       Qe = QHj + (size_t)e * HD;
  float acc[HEADS] = {0.f, 0.f, 0.f, 0.f};
  #pragma unroll
  for (int j = 0; j < 8; ++j) {
    const int   idx = j * 32 + lane;      // all 32 lanes of one j share head j>>1
    const int   h   = j >> 1;
    const int   d   = idx & 63;
    const float q   = Qe[idx];
    const float hi  = sp_f(Ps[idx] + q);
    const float hj  = sp_f(Pd[idx] + q);
    Qe[idx] = hj;
    acc[h] += hi * att[h * 128 + d] + hj * att[h * 128 + 64 + d];
  }
  #pragma unroll
  for (int h = 0; h < HEADS; ++h) {
    float v = acc[h];
    for (int off = 16; off > 0; off >>= 1) v += __shfl_xor(v, off, 32);
    if (lane == 0) alpha_raw[(size_t)e * HEADS + h] = sp_f(v);
  }
}

// BatchNorm stats over edges (per head): sums + sumsq into stats[0..7]
__global__ void alpha_stats(const float* __restrict__ alpha, float* __restrict__ stats)
{
  float s[HEADS] = {0, 0, 0, 0}, q[HEADS] = {0, 0, 0, 0};
  long i = (long)blockIdx.x * blockDim.x + threadIdx.x;
  const long stride = (long)gridDim.x * blockDim.x;
  for (long e = i; e < NEDGES; e += stride) {
    #pragma unroll
    for (int h = 0; h < HEADS; ++h) {
      const float v = alpha[e * HEADS + h];
      s[h] += v; q[h] += v * v;
    }
  }
  #pragma unroll
  for (int h = 0; h < HEADS; ++h) {
    atomicAdd(&stats[h], s[h]);
    atomicAdd(&stats[HEADS + h], q[h]);
  }
}

// alpha <- sp(BN(alpha)); also segment-max keyed by src via uint atomicMax
// (values are softplus outputs > 0, so IEEE bits are order-preserving; smax
// buffer is pre-zeroed = +0.0f bits).
__global__ void alpha_bn_apply(float* __restrict__ alpha, const float* __restrict__ stats,
                               const float* __restrict__ g, const float* __restrict__ b,
                               const int* __restrict__ src, float* __restrict__ smax)
{
  const long i = (long)blockIdx.x * blockDim.x + threadIdx.x;
  if (i >= (long)NEDGES * HEADS) return;
  const int  h    = (int)(i & 3);
  const long e    = i >> 2;
  const float invE = 1.f / (float)NEDGES;
  const float mean = stats[h] * invE;
  const float var  = stats[HEADS + h] * invE - mean * mean;
  const float v = sp_f((alpha[i] - mean) * rsqrtf(var + 1e-5f) * g[h] + b[h]);
  alpha[i] = v;
  atomicMax((unsigned int*)&smax[(size_t)src[e] * HEADS + h], __float_as_uint(v));
}

// alpha <- exp(alpha - smax[src]); segment-sum into ssum
__global__ void alpha_exp_sum(float* __restrict__ alpha, const float* __restrict__ smax,
                              float* __restrict__ ssum, const int* __restrict__ src)
{
  const long i = (long)blockIdx.x * blockDim.x + threadIdx.x;
  if (i >= (long)NEDGES * HEADS) return;
  const int  h  = (int)(i & 3);
  const long e  = i >> 2;
  const int  sn = src[e];
  const float ex = expf(alpha[i] - smax[(size_t)sn * HEADS + h]);
  alpha[i] = ex;
  atomicAdd(&ssum[(size_t)sn * HEADS + h], ex);
}

// aggr[src] += h_j * alpha/(ssum+eps). One wave per edge, f32 atomic adds.
__global__ void edge_aggr(const float* __restrict__ Hj, const float* __restrict__ alpha,
                          const float* __restrict__ ssum, const int* __restrict__ src,
                          float* __restrict__ aggr)
{
  const int e    = (int)((blockIdx.x * blockDim.x + threadIdx.x) >> 5);
  const int lane = threadIdx.x & 31;
  if (e >= NEDGES) return;
  const int sn = src[e];
  float coef[HEADS];
  #pragma unroll
  for (int h = 0; h < HEADS; ++h)
    coef[h] = alpha[(size_t)e * HEADS + h] / (ssum[(size_t)sn * HEADS + h] + 1e-16f);
  #pragma unroll
  for (int j = 0; j < 8; ++j) {
    const int idx = j * 32 + lane;
    atomicAdd(&aggr[(size_t)sn * HD + idx], Hj[(size_t)e * HD + idx] * coef[j >> 1]);
  }
}

// h = mean over heads + conv_bias
__global__ void node_h(const float* __restrict__ aggr, const float* __restrict__ bias,
                       float* __restrict__ h)
{
  const int i = blockIdx.x * blockDim.x + threadIdx.x;
  if (i >= NNODES * DIM) return;
  const int n = i >> 6, d = i & 63;
  const float* a = aggr + (size_t)n * HD;
  h[i] = 0.25f * (a[d] + a[64 + d] + a[128 + d] + a[192 + d]) + bias[d];
}

// s = softmax(h @ dgn_lw + dgn_lb) over GROUPS
__global__ void dgn_softmax(const float* __restrict__ h, const float* __restrict__ lw,
                            const float* __restrict__ lb, float* __restrict__ sOut)
{
  const int n = blockIdx.x * blockDim.x + threadIdx.x;
  if (n >= NNODES) return;
  float lg[GROUPS];
  #pragma unroll
  for (int g = 0; g < GROUPS; ++g) lg[g] = lb[g];
  const float* hn = h + (size_t)n * DIM;
  for (int d = 0; d < DIM; ++d) {
    const float hv = hn[d];
    #pragma unroll
    for (int g = 0; g < GROUPS; ++g) lg[g] += hv * lw[d * GROUPS + g];
  }
  float m = lg[0];
  #pragma unroll
  for (int g = 1; g < GROUPS; ++g) m = fmaxf(m, lg[g]);
  float ss = 0.f;
  #pragma unroll
  for (int g = 0; g < GROUPS; ++g) { lg[g] = expf(lg[g] - m); ss += lg[g]; }
  const float inv = 1.f / ss;
  #pragma unroll
  for (int g = 0; g < GROUPS; ++g) sOut[(size_t)n * GROUPS + g] = lg[g] * inv;
}

// Column mean/rstd of o[n, g*64+d] = s[n,g]*h[n,d] over nodes. Block per column.
__global__ void dgn_colstats(const float* __restrict__ h, const float* __restrict__ s,
                             float* __restrict__ cmean, float* __restrict__ crstd)
{
  const int c = blockIdx.x, g = c >> 6, d = c & 63;
  float su = 0.f, sq = 0.f;
  for (int n = threadIdx.x; n < NNODES; n += blockDim.x) {
    const float o = s[(size_t)n * GROUPS + g] * h[(size_t)n * DIM + d];
    su += o; sq += o * o;
  }
  __shared__ float r1[256], r2[256];
  r1[threadIdx.x] = su; r2[threadIdx.x] = sq; __syncthreads();
  for (int t = 128; t > 0; t >>= 1) {
    if ((int)threadIdx.x < t) { r1[threadIdx.x] += r1[threadIdx.x + t]; r2[threadIdx.x] += r2[threadIdx.x + t]; }
    __syncthreads();
  }
  if (threadIdx.x == 0) {
    const float mean = r1[0] / (float)NNODES;
    const float var  = r2[0] / (float)NNODES - mean * mean;
    cmean[c] = mean;
    crstd[c] = rsqrtf(var + 1e-5f);
  }
}

// out_x += h + 0.01 * sum_g normalized(o); (prev residual is current out_x)
__global__ void dgn_update(float* __restrict__ out_x, const float* __restrict__ h,
                           const float* __restrict__ s, const float* __restrict__ cmean,
                           const float* __restrict__ crstd, const float* __restrict__ gg,
                           const float* __restrict__ gb)
{
  const int i = blockIdx.x * blockDim.x + threadIdx.x;
  if (i >= NNODES * DIM) return;
  const int n = i >> 6, d = i & 63;
  const float hv = h[i];
  float accu = 0.f;
  #pragma unroll
  for (int g = 0; g < GROUPS; ++g) {
    const int   c = g * 64 + d;
    const float o = s[(size_t)n * GROUPS + g] * hv;
    accu += (o - cmean[c]) * crstd[c] * gg[c] + gb[c];
  }
  out_x[i] = out_x[i] + hv + 0.01f * accu;
}

// concat(out_x, glbl_x) -> [N, 172]
__global__ void cat_build(const float* __restrict__ out_x, const float* __restrict__ glbl,
                          float* __restrict__ cat)
{
  const int i = blockIdx.x * blockDim.x + threadIdx.x;
  if (i >= NNODES * 172) return;
  const int n = i / 172, j = i - n * 172;
  cat[i] = (j < DIM) ? out_x[(size_t)n * DIM + j] : glbl[(size_t)n * 108 + (j - DIM)];
}

// gsc[n] = g2[n] @ ga_w2 + b2
__global__ void gsc_dot(const float* __restrict__ g2, const float* __restrict__ w,
                        const float* __restrict__ b, float* __restrict__ gsc)
{
  const int n = blockIdx.x * blockDim.x + threadIdx.x;
  if (n >= NNODES) return;
  float s = b[0];
  const float* gn = g2 + (size_t)n * DIM;
  for (int k = 0; k < DIM; ++k) s += gn[k] * w[k];
  gsc[n] = s;
}

__global__ void graph_max(const float* __restrict__ gsc, const int* __restrict__ batch,
                          float* __restrict__ gm)
{
  const int b = blockIdx.x;
  float m = -INFINITY;
  for (int n = threadIdx.x; n < NNODES; n += blockDim.x)
    if (batch[n] == b) m = fmaxf(m, gsc[n]);
  __shared__ float red[256];
  red[threadIdx.x] = m; __syncthreads();
  for (int t = 128; t > 0; t >>= 1) {
    if ((int)threadIdx.x < t) red[threadIdx.x] = fmaxf(red[threadIdx.x], red[threadIdx.x + t]);
    __syncthreads();
  }
  if (threadIdx.x == 0) gm[b] = red[0];
}

__global__ void graph_sum(const float* __restrict__ gsc, const int* __restrict__ batch,
                          const float* __restrict__ gm, float* __restrict__ gs)
{
  const int b = blockIdx.x;
  float s = 0.f;
  for (int n = threadIdx.x; n < NNODES; n += blockDim.x)
    if (batch[n] == b) s += expf(gsc[n] - gm[b]);
  __shared__ float red[256];
  red[threadIdx.x] = s; __syncthreads();
  for (int t = 128; t > 0; t >>= 1) {
    if ((int)threadIdx.x < t) red[threadIdx.x] += red[threadIdx.x + t];
    __syncthreads();
  }
  if (threadIdx.x == 0) gs[b] = red[0];
}

__global__ void pool_scatter(const float* __restrict__ out_x, const float* __restrict__ gsc,
                             const float* __restrict__ gm, const float* __restrict__ gs,
                             const int* __restrict__ batch, float* __restrict__ pooled)
{
  const int i = blockIdx.x * blockDim.x + threadIdx.x;
  if (i >= NNODES * DIM) return;
  const int n = i >> 6, d = i & 63;
  const int b = batch[n];
  const float a = expf(gsc[n] - gm[b]) / (gs[b] + 1e-16f);
  atomicAdd(&pooled[(size_t)b * DIM + d], out_x[i] * a);
}

__global__ void final_out(const float* __restrict__ hf, const float* __restrict__ w,
                          const float* __restrict__ b, float* __restrict__ out)
{
  const int g = threadIdx.x;
  if (g >= NGRAPH) return;
  float s = b[0];
  const float* hg = hf + (size_t)g * DIM;
  for (int k = 0; k < DIM; ++k) s += hg[k] * w[k];
  out[g] = s;
}

// ---------------------------------------------------------------------------
static inline void launch_gemm(const float* A, int lda, const float* B, int ldb,
                               float* C, int ldc, int M, int K, int Nn,
                               const float* bias, int act, hipStream_t st)
{
  const int waves   = (M / 16) * (Nn / 64);
  const int threads = 256;
  const int blocks  = (waves * 32 + threads - 1) / threads;
  gemm_wmma_f32<<<blocks, threads, 0, st>>>(A, lda, B, ldb, C, ldc, M, K, Nn, bias, act);
}

extern "C" void kernel_launch(void* const* d_in, const int* in_sizes, int n_in,
                              void* d_out, int out_size, void* d_ws, size_t ws_size,
                              hipStream_t stream)
{
  (void)in_sizes; (void)n_in; (void)out_size; (void)ws_size;
  const float* x         = (const float*)d_in[0];
  const float* edge_attr = (const float*)d_in[1];
  const float* glbl_x    = (const float*)d_in[2];
  const int*   eidx      = (const int*)  d_in[3];
  const int*   batch     = (const int*)  d_in[4];
  const float* pre_w_n   = (const float*)d_in[5];
  const float* pre_b_n   = (const float*)d_in[6];
  const float* pre_w_e   = (const float*)d_in[7];
  const float* pre_b_e   = (const float*)d_in[8];
  const float* conv_W    = (const float*)d_in[9];
  const float* conv_att  = (const float*)d_in[10];
  const float* conv_bias = (const float*)d_in[11];
  const float* bn1_g     = (const float*)d_in[12];
  const float* bn1_b     = (const float*)d_in[13];
  const float* dgn_lw    = (const float*)d_in[14];
  const float* dgn_lb    = (const float*)d_in[15];
  const float* dgn_g     = (const float*)d_in[16];
  const float* dgn_b     = (const float*)d_in[17];
  const float* ga_w0 = (const float*)d_in[18]; const float* ga_b0 = (const float*)d_in[19];
  const float* ga_w1 = (const float*)d_in[20]; const float* ga_b1 = (const float*)d_in[21];
  const float* ga_w2 = (const float*)d_in[22]; const float* ga_b2 = (const float*)d_in[23];
  const float* post_w = (const float*)d_in[24]; const float* post_b = (const float*)d_in[25];
  const float* out_w  = (const float*)d_in[26]; const float* out_b  = (const float*)d_in[27];

  const int* src = eidx;            // edge_index[0] — aggregation target
  const int* dst = eidx + NEDGES;   // edge_index[1] — message source

  // Workspace layout (floats). stats..aggr contiguous for one-shot zeroing.
  float* ws    = (float*)d_ws;
  float* out_x = ws;                      // 640,000
  float* out_e = out_x + 640000;          // 10,240,000
  float* P     = out_e + 10240000;        // 2,560,000
  float* Q     = P     + 2560000;         // 40,960,000 (also h_j, post-layer arena)
  float* alpha = Q     + 40960000;        // 640,000
  float* stats = alpha + 640000;          // 16
  float* smax  = stats + 16;              // 40,000
  float* ssum  = smax  + 40000;           // 40,000
  float* aggr  = ssum  + 40000;           // 2,560,000
  float* hbuf  = aggr  + 2560000;         // 640,000
  float* sbuf  = hbuf  + 640000;          // 100,000
  float* cmean = sbuf  + 100000;          // 640
  float* crstd = cmean + 640;             // 640
  // Post-layer buffers alias the Q arena (Q dead after the last layer):
  float* catb   = Q;                      // 1,720,000
  float* g1     = catb + 1720000;         // 640,000
  float* g2     = g1   + 640000;          // 640,000
  float* gsc    = g2   + 640000;          // 10,000
  float* gm     = gsc  + 10000;           // 64
  float* gs     = gm   + 64;              // 64
  float* pooled = gs   + 64;              // 4,096
  float* hfin   = pooled + 4096;          // 4,096

  // Pre-layers (leaky_relu 0.2)
  launch_gemm(x,         92, pre_w_n, DIM, out_x, DIM, NNODES, 92, DIM, pre_b_n, 1, stream);
  launch_gemm(edge_attr, 50, pre_w_e, DIM, out_e, DIM, NEDGES, 50, DIM, pre_b_e, 1, stream);

  for (int l = 0; l < NLAYER; ++l) {
    const float* Wl = conv_W + (size_t)l * 128 * HD;
    // P = out_x @ W[:64], Q = out_e @ W[64:]
    launch_gemm(out_x, DIM, Wl,                    HD, P, HD, NNODES, DIM, HD, nullptr, 0, stream);
    launch_gemm(out_e, DIM, Wl + (size_t)64 * HD,  HD, Q, HD, NEDGES, DIM, HD, nullptr, 0, stream);
    // zero stats(16)+smax(40k)+ssum(40k)+aggr(2.56M) in one pass
    zero_f32<<<2048, 256, 0, stream>>>(stats, 16L + 40000 + 40000 + 2560000);
    edge_msg<<<NEDGES / 8, 256, 0, stream>>>(P, Q, src, dst,
                                             conv_att + (size_t)l * HEADS * 128, alpha);
    alpha_stats<<<256, 256, 0, stream>>>(alpha, stats);
    alpha_bn_apply<<<(NEDGES * HEADS + 255) / 256, 256, 0, stream>>>(
        alpha, stats, bn1_g + l * HEADS, bn1_b + l * HEADS, src, smax);
    alpha_exp_sum<<<(NEDGES * HEADS + 255) / 256, 256, 0, stream>>>(alpha, smax, ssum, src);
    edge_aggr<<<NEDGES / 8, 256, 0, stream>>>(Q, alpha, ssum, src, aggr);
    node_h<<<(NNODES * DIM + 255) / 256, 256, 0, stream>>>(aggr, conv_bias + l * DIM, hbuf);
    dgn_softmax<<<(NNODES + 255) / 256, 256, 0, stream>>>(
        hbuf, dgn_lw + (size_t)l * DIM * GROUPS, dgn_lb + l * GROUPS, sbuf);
    dgn_colstats<<<GROUPS * DIM, 256, 0, stream>>>(hbuf, sbuf, cmean, crstd);
    dgn_update<<<(NNODES * DIM + 255) / 256, 256, 0, stream>>>(
        out_x, hbuf, sbuf, cmean, crstd,
        dgn_g + (size_t)l * GROUPS * DIM, dgn_b + (size_t)l * GROUPS * DIM);
  }

  // Global attention pooling + head
  cat_build<<<(NNODES * 172 + 255) / 256, 256, 0, stream>>>(out_x, glbl_x, catb);
  launch_gemm(catb, 172, ga_w0, DIM, g1, DIM, NNODES, 172, DIM, ga_b0, 2, stream);
  launch_gemm(g1,   DIM, ga_w1, DIM, g2, DIM, NNODES, DIM, DIM, ga_b1, 2, stream);
  gsc_dot<<<(NNODES + 255) / 256, 256, 0, stream>>>(g2, ga_w2, ga_b2, gsc);
  graph_max<<<NGRAPH, 256, 0, stream>>>(gsc, batch, gm);
  graph_sum<<<NGRAPH, 256, 0, stream>>>(gsc, batch, gm, gs);
  zero_f32<<<16, 256, 0, stream>>>(pooled, NGRAPH * DIM);
  pool_scatter<<<(NNODES * DIM + 255) / 256, 256, 0, stream>>>(out_x, gsc, gm, gs, batch, pooled);
  launch_gemm(pooled, DIM, post_w, DIM, hfin, DIM, NGRAPH, DIM, DIM, post_b, 2, stream);
  final_out<<<1, 64, 0, stream>>>(hfin, out_w, out_b, (float*)d_out);
}